// LSTMModel_3229815407187
// MI455X (gfx1250) — compile-verified
//
#include <hip/hip_runtime.h>

// LSTM stack for MI455X (gfx1250, wave32).
// Layer-by-layer scan; one wave (32 threads) per 16-batch tile.
// Gates via V_WMMA_F32_16X16X4_F32 (exact fp32 CDNA5 matrix path).
// h_seq ping-pongs through d_ws (2 x 84 MB -> resident in 192 MB global L2).
// Steady-state T-loop: coalesced loads -> 18 WMMA -> LDS scatter -> branch-free
// fast sigmoid/tanh (v_rcp_f32 + v_exp_f32) -> h written straight back into the
// WMMA A-tile; c-state lives in registers.

#define B_TOT 4096
#define T_LEN 512
#define HID   10
#define NL    50
#define KP    24   // padded K (max in+H = 22 -> 24, 6 K-steps of 4)
#define NP    48   // padded gate dim (40 -> 3 N-tiles of 16)

typedef __attribute__((ext_vector_type(2))) float v2f;
typedef __attribute__((ext_vector_type(8))) float v8f;

__device__ __forceinline__ float rcp_(float x)  { return __builtin_amdgcn_rcpf(x); }
__device__ __forceinline__ float sig_(float x)  { return rcp_(1.0f + __expf(-x)); }
__device__ __forceinline__ float tanh_(float x) {
    // tanh(x) = 1 - 2/(exp(2x)+1): trans + fma only, no divide expansion
    return __builtin_fmaf(-2.0f, rcp_(1.0f + __expf(2.0f * x)), 1.0f);
}

// INF: input features (12 layer 0, 10 after). XLAYOUT: x[B,T,INF] vs h_seq[T,B,INF].
template <int INF, bool XLAYOUT>
__global__ void __launch_bounds__(32)
lstm_layer_kernel(const float* __restrict__ in,
                  const float* __restrict__ W_ih,   // [40, INF]
                  const float* __restrict__ W_hh,   // [40, 10]
                  const float* __restrict__ b_ih,   // [40]
                  const float* __restrict__ b_hh,   // [40]
                  float* __restrict__ h_out)        // [T, B, 10]
{
    __shared__ float sW[KP][NP];   // Wcat[k][n], zero padded
    __shared__ float sB[NP];       // combined bias
    __shared__ float sA[16][KP];   // A tile: [row][k] = concat(input, h_state)
    __shared__ float sG[16][NP];   // gates redistributed from WMMA C layout

    const int lane = threadIdx.x;      // 0..31
    const int half = lane >> 4;        // K-pair select
    const int mrow = lane & 15;        // A row / C column
    const int b0   = blockIdx.x * 16;

    // ---- setup: weights/bias -> LDS, zero A tile (pad + initial h state) ----
    for (int e = lane; e < KP * NP; e += 32) {
        int k = e / NP, n = e % NP;
        float w = 0.0f;
        if (n < 40) {
            if (k < INF)            w = W_ih[n * INF + k];
            else if (k < INF + HID) w = W_hh[n * HID + (k - INF)];
        }
        sW[k][n] = w;
    }
    for (int n = lane; n < NP; n += 32) sB[n] = (n < 40) ? (b_ih[n] + b_hh[n]) : 0.0f;
    for (int e = lane; e < 16 * KP; e += 32) sA[e / KP][e % KP] = 0.0f;
    __syncthreads();

    // ---- hoist B-matrix fragments + per-column bias into registers ----
    v2f bw[6][3];
#pragma unroll
    for (int kk = 0; kk < 6; ++kk) {
        const int k0 = kk * 4 + half * 2;
#pragma unroll
        for (int nt = 0; nt < 3; ++nt) {
            const int n = nt * 16 + mrow;
            v2f b; b.x = sW[k0][n]; b.y = sW[k0 + 1][n];
            bw[kk][nt] = b;
        }
    }
    float bias[3];
#pragma unroll
    for (int nt = 0; nt < 3; ++nt) bias[nt] = sB[nt * 16 + mrow];

    // ---- precomputed per-lane index maps (constant over T) ----
    constexpr int NU = (16 * INF) / 32;      // input elements per lane (6 or 5)
    int li_m[NU], li_i[NU];
    size_t li_g[NU];
#pragma unroll
    for (int u = 0; u < NU; ++u) {
        const int e = lane + 32 * u;
        const int m = e / INF, i = e % INF;
        li_m[u] = m; li_i[u] = i;
        li_g[u] = XLAYOUT ? ((size_t)(b0 + m)) * T_LEN * INF + i
                          : (size_t)(b0 * INF + e);
    }
    int um[5], uj[5], uo[5];                 // elementwise units: 5 per lane
    float creg[5];
#pragma unroll
    for (int u = 0; u < 5; ++u) {
        const int p = lane + 32 * u;
        um[u] = p / HID; uj[u] = p % HID; uo[u] = um[u] * HID + uj[u];
        creg[u] = 0.0f;
    }

    // ---- scan over time ----
    for (int t = 0; t < T_LEN; ++t) {
        const size_t tstep = XLAYOUT ? (size_t)t * INF : (size_t)t * B_TOT * INF;
        // input slice -> A tile (coalesced for h layout)
#pragma unroll
        for (int u = 0; u < NU; ++u)
            sA[li_m[u]][li_i[u]] = in[li_g[u] + tstep];
        if (t + 1 < T_LEN)   // global_prefetch_b8 for next timestep
            __builtin_prefetch(&in[li_g[0] + tstep + (XLAYOUT ? INF : B_TOT * INF)], 0, 3);
        __syncthreads();

        // gates = A @ Wcat + bias : 6 K-steps x 3 N-tiles of WMMA f32 16x16x4
        v8f acc0, acc1, acc2;
#pragma unroll
        for (int r = 0; r < 8; ++r) { acc0[r] = bias[0]; acc1[r] = bias[1]; acc2[r] = bias[2]; }
#pragma unroll
        for (int kk = 0; kk < 6; ++kk) {
            const int k0 = kk * 4 + half * 2;
            v2f a; a.x = sA[mrow][k0]; a.y = sA[mrow][k0 + 1];
            acc0 = __builtin_amdgcn_wmma_f32_16x16x4_f32(false, a, false, bw[kk][0],
                                                         (short)0, acc0, false, false);
            acc1 = __builtin_amdgcn_wmma_f32_16x16x4_f32(false, a, false, bw[kk][1],
                                                         (short)0, acc1, false, false);
            acc2 = __builtin_amdgcn_wmma_f32_16x16x4_f32(false, a, false, bw[kk][2],
                                                         (short)0, acc2, false, false);
        }
        // scatter C (VGPR r -> row half*8+r, col lane&15) into sG
#pragma unroll
        for (int r = 0; r < 8; ++r) {
            const int m2 = half * 8 + r;
            sG[m2][0 * 16 + mrow] = acc0[r];
            sG[m2][1 * 16 + mrow] = acc1[r];
            sG[m2][2 * 16 + mrow] = acc2[r];
        }
        __syncthreads();

        // LSTM cell: 5 units/lane, c in registers, h -> A tile + global
        const size_t obase = (size_t)t * B_TOT * HID + (size_t)b0 * HID;
#pragma unroll
        for (int u = 0; u < 5; ++u) {
            const int m = um[u], j = uj[u];
            const float ig = sG[m][j];
            const float fg = sG[m][HID + j];
            const float gg = sG[m][2 * HID + j];
            const float og = sG[m][3 * HID + j];
            const float c  = sig_(fg) * creg[u] + sig_(ig) * tanh_(gg);
            const float h  = sig_(og) * tanh_(c);
            creg[u] = c;
            sA[m][INF + j] = h;                  // next timestep's recurrent input
            h_out[obase + uo[u]] = h;            // coalesced 640B per wave
        }
        __syncthreads();
    }
}

__global__ void __launch_bounds__(256)
fc_sigmoid_kernel(const float* __restrict__ h,      // [T, B, 10]
                  const float* __restrict__ W_fc,   // [1, 10]
                  const float* __restrict__ b_fc,   // [1]
                  float* __restrict__ out)          // [B, T]
{
    const size_t e = (size_t)blockIdx.x * blockDim.x + threadIdx.x;
    if (e >= (size_t)B_TOT * T_LEN) return;
    const int t = (int)(e / B_TOT);
    const int b = (int)(e % B_TOT);
    const float* hp = h + (((size_t)t) * B_TOT + b) * HID;
    float acc = b_fc[0];
#pragma unroll
    for (int j = 0; j < HID; ++j) acc += hp[j] * W_fc[j];
    out[(size_t)b * T_LEN + t] = sig_(acc);
}

extern "C" void kernel_launch(void* const* d_in, const int* in_sizes, int n_in,
                              void* d_out, int out_size, void* d_ws, size_t ws_size,
                              hipStream_t stream) {
    const float* x      = (const float*)d_in[0];
    const float* W_ih0  = (const float*)d_in[1];
    const float* W_hh0  = (const float*)d_in[2];
    const float* b_ih0  = (const float*)d_in[3];
    const float* b_hh0  = (const float*)d_in[4];
    const float* W_ih_r = (const float*)d_in[5];   // [49, 40, 10]
    const float* W_hh_r = (const float*)d_in[6];   // [49, 40, 10]
    const float* b_ih_r = (const float*)d_in[7];   // [49, 40]
    const float* b_hh_r = (const float*)d_in[8];   // [49, 40]
    const float* W_fc   = (const float*)d_in[9];
    const float* b_fc   = (const float*)d_in[10];
    float* out = (float*)d_out;

    float* bufA = (float*)d_ws;                           // 84 MB
    float* bufB = bufA + (size_t)T_LEN * B_TOT * HID;     // 84 MB

    dim3 grid(B_TOT / 16), block(32);
    lstm_layer_kernel<12, true><<<grid, block, 0, stream>>>(
        x, W_ih0, W_hh0, b_ih0, b_hh0, bufA);

    const float* cur = bufA;
    float* nxt = bufB;
    for (int l = 1; l < NL; ++l) {
        lstm_layer_kernel<10, false><<<grid, block, 0, stream>>>(
            cur,
            W_ih_r + (size_t)(l - 1) * 40 * HID,
            W_hh_r + (size_t)(l - 1) * 40 * HID,
            b_ih_r + (size_t)(l - 1) * 40,
            b_hh_r + (size_t)(l - 1) * 40,
            nxt);
        float* tmp = (float*)cur; cur = nxt; nxt = tmp;
    }

    const int total = B_TOT * T_LEN;
    fc_sigmoid_kernel<<<(total + 255) / 256, 256, 0, stream>>>(cur, W_fc, b_fc, out);
}